// GraphSAGE_12189117186935
// MI455X (gfx1250) — compile-verified
//
#include <hip/hip_runtime.h>

typedef float v2f __attribute__((ext_vector_type(2)));
typedef float v8f __attribute__((ext_vector_type(8)));

// ---------------------------------------------------------------------------
// CDNA5 async global->LDS copy (ASYNCcnt path), with safe fallback.
// Builtin signature (from compiler diagnostic): arg0 = global (AS1) pointer to
// 16-byte int vector, arg1 = LDS (AS3) pointer, then offset + cpol ints.
// ---------------------------------------------------------------------------
#if __has_builtin(__builtin_amdgcn_global_load_async_to_lds_b128)
#define SAGE_ASYNC_LDS 1
#endif

typedef int b128_t __attribute__((vector_size(16)));
typedef __attribute__((address_space(1))) b128_t* gptr128_t;
typedef __attribute__((address_space(3))) b128_t* lptr128_t;

__device__ __forceinline__ void cp16_to_lds(const float* g, float* l) {
#ifdef SAGE_ASYNC_LDS
    __builtin_amdgcn_global_load_async_to_lds_b128((gptr128_t)g, (lptr128_t)l,
                                                   0, 0);
#else
    *(float4*)l = *(const float4*)g;
#endif
}

__device__ __forceinline__ void cp_wait_all() {
#ifdef SAGE_ASYNC_LDS
#if __has_builtin(__builtin_amdgcn_s_wait_asynccnt)
    __builtin_amdgcn_s_wait_asynccnt(0);
#else
    asm volatile("s_wait_asynccnt 0" ::: "memory");
#endif
#endif
}

// ---------------------------------------------------------------------------
// Utility kernels
// ---------------------------------------------------------------------------

__global__ void zero_f32_kernel(float* __restrict__ p, int n) {
    int i = blockIdx.x * blockDim.x + threadIdx.x;
    if (i < n) p[i] = 0.0f;
}

__global__ void count_deg_kernel(const long long* __restrict__ eidx,
                                 float* __restrict__ deg, int E) {
    int e = blockIdx.x * blockDim.x + threadIdx.x;
    if (e < E) {
        int dst = (int)eidx[(long long)E + e];
        atomicAdd(&deg[dst], 1.0f);
    }
}

__global__ void invdeg_kernel(float* __restrict__ deg, int n) {
    int i = blockIdx.x * blockDim.x + threadIdx.x;
    if (i < n) deg[i] = 1.0f / fmaxf(deg[i], 1.0f);
}

// Zero-pad layer-5 weights 64x17 -> 64x32 and bias 17 -> 32.
__global__ void pack_w5_kernel(const float* __restrict__ Wl,
                               const float* __restrict__ Wr,
                               const float* __restrict__ b,
                               float* __restrict__ Wlp,
                               float* __restrict__ Wrp,
                               float* __restrict__ bp) {
    int i = blockIdx.x * blockDim.x + threadIdx.x;
    if (i < 64 * 32) {
        int r = i >> 5, c = i & 31;
        Wlp[i] = (c < 17) ? Wl[r * 17 + c] : 0.0f;
        Wrp[i] = (c < 17) ? Wr[r * 17 + c] : 0.0f;
    }
    if (i < 32) bp[i] = (i < 17) ? b[i] : 0.0f;
}

// ---------------------------------------------------------------------------
// Edge scatter: msg[dst] += x[src].  16 lanes per edge, float4 per lane.
// Entire working set (x, msg, edges) is L2-resident; atomics execute at L2.
// ---------------------------------------------------------------------------

__global__ void scatter_kernel(const long long* __restrict__ eidx,
                               const float* __restrict__ x,
                               float* __restrict__ msg, int E) {
    int t = blockIdx.x * blockDim.x + threadIdx.x;   // E*16 = 25.6M fits in int
    if (t >= E * 16) return;
    int e = t >> 4;
    int c = (t & 15) << 2;                           // feature chunk of 4
    int s = (int)eidx[e];
    int d = (int)eidx[(long long)E + e];
    const float4 v = *(const float4*)(x + (size_t)s * 64 + c);
    float* m = msg + (size_t)d * 64 + c;
    atomicAdd(m + 0, v.x);
    atomicAdd(m + 1, v.y);
    atomicAdd(m + 2, v.z);
    atomicAdd(m + 3, v.w);
}

// ---------------------------------------------------------------------------
// Fused SAGE linear: out = relu( (msg * invdeg) @ Wl + bias + x @ Wr )
// One wave = one 16x16 output tile via V_WMMA_F32_16X16X4_F32.
// A tiles + W panels staged to LDS (async-to-LDS when available), with
// bank-conflict-free padded layouts.
// ---------------------------------------------------------------------------

template <int NTILES, bool STORE17>
__global__ __launch_bounds__(32 * NTILES)
void sage_wmma_kernel(const float* __restrict__ msg,
                      const float* __restrict__ invdeg,
                      const float* __restrict__ x,
                      const float* __restrict__ Wl,   // [64 x 16*NTILES] row-major
                      const float* __restrict__ Wr,   // [64 x 16*NTILES] row-major
                      const float* __restrict__ bias, // [16*NTILES]
                      float* __restrict__ out) {
    constexpr int LDW = 16 * NTILES;   // W row width
    constexpr int AS  = 68;            // A-tile LDS row stride (banks 4r+k, 16B rows)
    constexpr int WS  = LDW + 16;      // W LDS row stride (half-waves hit disjoint banks)

    __shared__ float sMsg[16 * AS];
    __shared__ float sX[16 * AS];
    __shared__ float sWl[64 * WS];
    __shared__ float sWr[64 * WS];

    const int tid   = threadIdx.x;
    const int NT    = 32 * NTILES;
    const int mtile = blockIdx.x;

    // ---- Stage A tiles: 16 rows x 64 floats each (rows contiguous in global)
    const float* gM = msg + (size_t)mtile * 16 * 64;
    const float* gX = x   + (size_t)mtile * 16 * 64;
#pragma unroll
    for (int i = tid; i < 256; i += NT) {          // 256 float4 units per tile
        const int r = i >> 4, c = (i & 15) << 2;
        cp16_to_lds(gM + r * 64 + c, sMsg + r * AS + c);
        cp16_to_lds(gX + r * 64 + c, sX   + r * AS + c);
    }
    // ---- Stage W panels: 64 rows x LDW floats
#pragma unroll
    for (int i = tid; i < 16 * LDW; i += NT) {     // 64*(LDW/4) float4 units
        const int r = i / (LDW / 4), c = (i % (LDW / 4)) << 2;
        cp16_to_lds(Wl + r * LDW + c, sWl + r * WS + c);
        cp16_to_lds(Wr + r * LDW + c, sWr + r * WS + c);
    }
    cp_wait_all();
    __syncthreads();

    const int lane  = tid & 31;
    const int ntile = tid >> 5;            // wave id = column tile
    const int lrow  = lane & 15;           // A fragment: M row within tile
    const int koff  = (lane >> 4) << 1;    // 0 (lanes 0-15) or 2 (lanes 16-31)
    const int colw  = ntile * 16 + lrow;   // B fragment: N column in W panel

    const float idg = invdeg[mtile * 16 + lrow];

    v8f acc = {};
#pragma unroll
    for (int k0 = 0; k0 < 64; k0 += 4) {
        const int kk = k0 + koff;
        v2f a, b;
        // mean-aggregated neighbors @ Wl   (1/deg fused into A fragment)
        a.x = sMsg[lrow * AS + kk]     * idg;
        a.y = sMsg[lrow * AS + kk + 1] * idg;
        b.x = sWl[kk * WS + colw];
        b.y = sWl[(kk + 1) * WS + colw];
        acc = __builtin_amdgcn_wmma_f32_16x16x4_f32(false, a, false, b,
                                                    (short)0, acc, false, false);
        // self features @ Wr
        a.x = sX[lrow * AS + kk];
        a.y = sX[lrow * AS + kk + 1];
        b.x = sWr[kk * WS + colw];
        b.y = sWr[(kk + 1) * WS + colw];
        acc = __builtin_amdgcn_wmma_f32_16x16x4_f32(false, a, false, b,
                                                    (short)0, acc, false, false);
    }

    // D mapping: VGPR r -> M = 16*mtile + r + 8*(lane>=16), N = colw.
    const float bn    = bias[colw];
    const int   mbase = mtile * 16 + ((lane >> 4) << 3);
#pragma unroll
    for (int r = 0; r < 8; ++r) {
        float v = fmaxf(acc[r] + bn, 0.0f);   // bias + ReLU fused
        const int m = mbase + r;
        if (STORE17) {
            if (colw < 17) out[(size_t)m * 17 + colw] = v;
        } else {
            out[(size_t)m * 64 + colw] = v;
        }
    }
}

// ---------------------------------------------------------------------------
// Host-side orchestration
// ---------------------------------------------------------------------------

extern "C" void kernel_launch(void* const* d_in, const int* in_sizes, int n_in,
                              void* d_out, int out_size, void* d_ws, size_t ws_size,
                              hipStream_t stream) {
    const float*     x    = (const float*)d_in[0];
    const long long* eidx = (const long long*)d_in[1];
    const float* Wl[5] = { (const float*)d_in[2],  (const float*)d_in[5],
                           (const float*)d_in[8],  (const float*)d_in[11],
                           (const float*)d_in[14] };
    const float* bl[5] = { (const float*)d_in[3],  (const float*)d_in[6],
                           (const float*)d_in[9],  (const float*)d_in[12],
                           (const float*)d_in[15] };
    const float* Wr[5] = { (const float*)d_in[4],  (const float*)d_in[7],
                           (const float*)d_in[10], (const float*)d_in[13],
                           (const float*)d_in[16] };

    const int N  = in_sizes[0] / 64;   // 100000
    const int E  = in_sizes[1] / 2;    // 1600000
    const int ND = N * 64;             // 6.4M floats

    // Workspace layout (floats): invdeg | msg | hA | hB | Wl5p | Wr5p | b5p
    float* ws     = (float*)d_ws;
    float* invdeg = ws;
    float* msg    = invdeg + N;
    float* hA     = msg + ND;
    float* hB     = hA + ND;
    float* Wl5p   = hB + ND;
    float* Wr5p   = Wl5p + 64 * 32;
    float* b5p    = Wr5p + 64 * 32;

    const int TB = 256;

    // Degree (recomputed every call; deterministic)
    zero_f32_kernel<<<(N + TB - 1) / TB, TB, 0, stream>>>(invdeg, N);
    count_deg_kernel<<<(E + TB - 1) / TB, TB, 0, stream>>>(eidx, invdeg, E);
    invdeg_kernel<<<(N + TB - 1) / TB, TB, 0, stream>>>(invdeg, N);

    // Pad layer-5 weights once per call
    pack_w5_kernel<<<(64 * 32 + TB - 1) / TB, TB, 0, stream>>>(
        Wl[4], Wr[4], bl[4], Wl5p, Wr5p, b5p);

    const float* cur = x;
    float* pp[2] = { hA, hB };
    const int mtiles = N / 16;                 // 6250, exact

    for (int l = 0; l < 5; ++l) {
        zero_f32_kernel<<<(ND + TB - 1) / TB, TB, 0, stream>>>(msg, ND);
        scatter_kernel<<<(E * 16 + TB - 1) / TB, TB, 0, stream>>>(eidx, cur, msg, E);
        if (l < 4) {
            float* nxt = pp[l & 1];
            sage_wmma_kernel<4, false><<<mtiles, 128, 0, stream>>>(
                msg, invdeg, cur, Wl[l], Wr[l], bl[l], nxt);
            cur = nxt;
        } else {
            sage_wmma_kernel<2, true><<<mtiles, 64, 0, stream>>>(
                msg, invdeg, cur, Wl5p, Wr5p, b5p, (float*)d_out);
        }
    }
}